// ValenceConstraint_61048665145612
// MI455X (gfx1250) — compile-verified
//
#include <hip/hip_runtime.h>

// ---------------------------------------------------------------------------
// ValenceConstraint on MI455X (gfx1250): pure bandwidth problem (~220 MB of
// traffic, ~25 MFLOP).  Floor = 220MB / 23.3 TB/s ~= 9.5 us.  No matmul =>
// no WMMA; chip-specific paths used: async Global->LDS DMA (ASYNCcnt),
// b128 NT loads/stores, global_prefetch_b8, wave32 blocks, deterministic
// u32 global atomics.
// ---------------------------------------------------------------------------

typedef float __attribute__((ext_vector_type(4))) vfloat4;

#define CONSTRAINT_WEIGHT 0.05f

// valence table: default 4, overrides for {1,6,7,8,9,15,16,17}.
__device__ __forceinline__ int valence_i(int z) {
    switch (z) {
        case 1:
        case 9:  return 1;
        case 8:  return 2;
        case 7:  return 3;
        case 15: return 5;
        case 16: return 6;
        case 17: return 7;
        default: return 4;   // covers 6 and every other (clipped) Z
    }
}

// --- kernel 1: zero counts[N] and vsum[1] (contiguous in ws) ----------------
__global__ void vc_zero_kernel(unsigned* __restrict__ p, int n) {
    int i = blockIdx.x * blockDim.x + threadIdx.x;
    if (i < n) p[i] = 0u;
}

// --- kernel 2: bond_counts via u32 scatter-add over edge sources ------------
__global__ void vc_count_edges_kernel(const int* __restrict__ row,
                                      unsigned* __restrict__ counts, int E) {
    int i = blockIdx.x * blockDim.x + threadIdx.x;
    if (i < E) {
        atomicAdd(&counts[row[i]], 1u);   // global_atomic_add_u32, no-return
    }
}

// --- kernel 3: integer violation sum (deterministic across replays) ---------
__global__ void vc_reduce_violation_kernel(const unsigned* __restrict__ counts,
                                           const int* __restrict__ types,
                                           unsigned* __restrict__ vsum, int N) {
    __shared__ unsigned sdata[256];
    int i = blockIdx.x * blockDim.x + threadIdx.x;
    unsigned v = 0u;
    if (i < N) {
        int d = (int)counts[i] - valence_i(types[i]);
        v = (d > 0) ? (unsigned)d : 0u;
    }
    sdata[threadIdx.x] = v;
    __syncthreads();
    #pragma unroll
    for (int s = 128; s > 0; s >>= 1) {
        if ((int)threadIdx.x < s) sdata[threadIdx.x] += sdata[threadIdx.x + s];
        __syncthreads();
    }
    if (threadIdx.x == 0) atomicAdd(vsum, sdata[0]);
}

// --- kernel 4: stream h -> h*penalty, staged through LDS via the gfx1250 ----
// async DMA path.  64 lanes * float4 = one 256-float row; 256 threads = 4 rows.
__global__ __launch_bounds__(256)
void vc_scale_rows_kernel(const float* __restrict__ h,
                          const unsigned* __restrict__ counts,
                          const int* __restrict__ types,
                          float* __restrict__ out, int N) {
    __shared__ __align__(16) float stage[256 * 4];    // 4 KB: one b128 per lane
    const int lanesPerRow  = 64;                      // D/4 = 256/4
    const int rowsPerBlock = 4;                       // 256 / 64
    int rowInBlock = threadIdx.x >> 6;
    int lane       = threadIdx.x & 63;
    int atom = blockIdx.x * rowsPerBlock + rowInBlock;
    if (atom >= N) return;

    // per-row scalar work: broadcast loads (same addr per wave, L2-resident)
    int d = (int)counts[atom] - valence_i(types[atom]);
    float penalty = (d > 0) ? fmaf((float)d, -0.1f, 1.0f) : 1.0f;

    size_t vidx = (size_t)atom * lanesPerRow + lane;
    const vfloat4* hp = (const vfloat4*)h + vidx;
    vfloat4*       op = (vfloat4*)out + vidx;

    // gfx1250 global_prefetch_b8 one grid-slice ahead (speculative => safe)
    __builtin_prefetch(hp + (size_t)gridDim.x * rowsPerBlock * lanesPerRow, 0, 0);

    // async Global->LDS b128: each lane DMAs its own 16B slot (ASYNCcnt path).
    // LDS byte address == low 32 bits of the generic pointer (aperture rule).
    vfloat4* sp = (vfloat4*)stage + threadIdx.x;
    unsigned lds_off = (unsigned)(size_t)sp;
    asm volatile("global_load_async_to_lds_b128 %0, %1, off"
                 :: "v"(lds_off), "v"(hp)
                 : "memory");
    asm volatile("s_wait_asynccnt 0x0" ::: "memory");

    vfloat4 v = *sp;            // ds_load_b128 from own slot (wave-local, no barrier)
    v *= penalty;
    __builtin_nontemporal_store(v, op);               // global_store_b128 th:NT
}

// --- kernel 5: loss = vsum * 0.05 / N ---------------------------------------
__global__ void vc_write_loss_kernel(const unsigned* __restrict__ vsum,
                                     float* __restrict__ out_loss, float invN) {
    if (blockIdx.x == 0 && threadIdx.x == 0) {
        *out_loss = (float)(*vsum) * CONSTRAINT_WEIGHT * invN;
    }
}

extern "C" void kernel_launch(void* const* d_in, const int* in_sizes, int n_in,
                              void* d_out, int out_size, void* d_ws, size_t ws_size,
                              hipStream_t stream) {
    (void)n_in; (void)ws_size; (void)out_size;

    const float* h     = (const float*)d_in[0];   // [N, 256] f32
    const int*   edges = (const int*)d_in[1];     // [2, E]; row 0 = first E
    const int*   types = (const int*)d_in[2];     // [N]

    const int N = in_sizes[2];
    const int E = in_sizes[1] / 2;
    const int D = in_sizes[0] / N;                // 256

    unsigned* counts = (unsigned*)d_ws;           // N u32
    unsigned* vsum   = counts + N;                // 1 u32
    float*    out    = (float*)d_out;             // N*D floats + 1 loss

    const int zn = N + 1;
    vc_zero_kernel<<<(zn + 255) / 256, 256, 0, stream>>>(counts, zn);

    vc_count_edges_kernel<<<(E + 255) / 256, 256, 0, stream>>>(edges, counts, E);

    vc_reduce_violation_kernel<<<(N + 255) / 256, 256, 0, stream>>>(counts, types,
                                                                    vsum, N);

    const int rowsPerBlock = 4;                   // 256 threads = 4 rows
    vc_scale_rows_kernel<<<(N + rowsPerBlock - 1) / rowsPerBlock, 256, 0, stream>>>(
        h, counts, types, out, N);

    vc_write_loss_kernel<<<1, 1, 0, stream>>>(vsum, out + (size_t)N * D,
                                              1.0f / (float)N);
}